// FeatureDiversity_49529562858364
// MI455X (gfx1250) — compile-verified
//
#include <hip/hip_runtime.h>
#include <hip/hip_bf16.h>

// CDNA5 / gfx1250: wave32. WMMA f32 16x16x4: A = 16x4 f32 (2 VGPRs),
// B = 4x16 f32 (2 VGPRs), C/D = 16x16 f32 (8 VGPRs).
typedef __attribute__((ext_vector_type(2))) float v2f;
typedef __attribute__((ext_vector_type(8))) float v8f;

#define FD_BLOCKS 2048
#define FD_WAVES_PER_BLOCK 8
#define FD_THREADS (FD_WAVES_PER_BLOCK * 32)

// Pass 1: bandwidth-bound row reduction.
// Each wave owns whole rows of N=1024 floats. Lane l loads float4 at
// (l + 32*j) -> every wave load instruction is 512 contiguous bytes
// (global_load_b128, fully coalesced). Per-lane s1 += x (v_add_f32),
// s2 = fma(x,x,s2). Then a 32-lane shfl_xor butterfly produces the row's
// s1/s2, and the nonlinear per-row term N*s2 - s1^2 accumulates per wave.
__global__ __launch_bounds__(FD_THREADS) void fd_rows_kernel(
    const float* __restrict__ x, float* __restrict__ partials, int R, int N) {
  const int lane = threadIdx.x & 31;
  const int wave = threadIdx.x >> 5;
  const int waveGid = blockIdx.x * FD_WAVES_PER_BLOCK + wave;
  const int totalWaves = gridDim.x * FD_WAVES_PER_BLOCK;

  float acc = 0.0f;
  for (int row = waveGid; row < R; row += totalWaves) {
    const float4* rp = (const float4*)(x + (size_t)row * (size_t)N);
    float s1 = 0.0f, s2 = 0.0f;
#pragma unroll
    for (int j = 0; j < 8; ++j) {  // 8 * 32 lanes * 4 floats = 1024 = N
      float4 v = rp[lane + 32 * j];
      s1 += v.x + v.y + v.z + v.w;
      s2 = fmaf(v.x, v.x, s2);
      s2 = fmaf(v.y, v.y, s2);
      s2 = fmaf(v.z, v.z, s2);
      s2 = fmaf(v.w, v.w, s2);
    }
    // wave32 butterfly reduction
#pragma unroll
    for (int off = 16; off >= 1; off >>= 1) {
      s1 += __shfl_xor(s1, off, 32);
      s2 += __shfl_xor(s2, off, 32);
    }
    acc += (float)N * s2 - s1 * s1;  // per-row pairwise term (unscaled)
  }

  __shared__ float lds[FD_WAVES_PER_BLOCK];
  if (lane == 0) lds[wave] = acc;
  __syncthreads();
  if (threadIdx.x == 0) {
    float t = 0.0f;
#pragma unroll
    for (int w = 0; w < FD_WAVES_PER_BLOCK; ++w) t += lds[w];
    partials[blockIdx.x] = t;
  }
}

// Pass 2: single wave, WMMA-based final reduction.
// A = ones(16x4). D[m,n] = sum_k B[k,n], so summing the 16 columns of D
// equals the sum of ALL 64 values held in B's two VGPRs — independent of
// the exact (k,n) slot mapping. Accumulate 64 partials per WMMA into C,
// then collapse the 16 columns (lanes n and n+16 hold column n) with a
// 4-step shfl_xor within each 16-lane half.
__global__ __launch_bounds__(32) void fd_reduce_kernel(
    const float* __restrict__ partials, float* __restrict__ out, int P,
    float scale) {
  const int lane = threadIdx.x;  // 0..31, EXEC all ones (no divergence here)
  v2f a;
  a[0] = 1.0f;
  a[1] = 1.0f;
  v8f c = {};  // zero accumulator

  const v2f* p2 = (const v2f*)partials;
  const int iters = P >> 6;  // 64 partials per WMMA
  for (int it = 0; it < iters; ++it) {
    v2f b = p2[it * 32 + lane];
    // v_wmma_f32_16x16x4_f32: (neg_a, A, neg_b, B, c_mod, C, reuse_a, reuse_b)
    c = __builtin_amdgcn_wmma_f32_16x16x4_f32(false, a, false, b, (short)0, c,
                                              false, false);
  }

  float t = c[0];  // lane holds column (lane % 16); rows identical (A=ones)
  t += __shfl_xor(t, 1, 32);
  t += __shfl_xor(t, 2, 32);
  t += __shfl_xor(t, 4, 32);
  t += __shfl_xor(t, 8, 32);  // lane 0: sum over 16 columns = total
  if (lane == 0) out[0] = t * scale;
}

extern "C" void kernel_launch(void* const* d_in, const int* in_sizes, int n_in,
                              void* d_out, int out_size, void* d_ws,
                              size_t ws_size, hipStream_t stream) {
  const float* x = (const float*)d_in[0];
  float* out = (float*)d_out;
  float* partials = (float*)d_ws;  // FD_BLOCKS floats of scratch

  const int N = 1024;                // last-axis length (reference shape)
  const int Bdim = 128;              // batch (reference shape)
  const int R = in_sizes[0] / N;     // 128*512 = 65536 rows

  fd_rows_kernel<<<FD_BLOCKS, FD_THREADS, 0, stream>>>(x, partials, R, N);

  // result = sum_rows(N*s2 - s1^2) / (N*(N-1)) / Bdim
  const float scale = 1.0f / ((float)N * (float)(N - 1) * (float)Bdim);
  fd_reduce_kernel<<<1, 32, 0, stream>>>(partials, out, FD_BLOCKS, scale);
}